// POELayer_1468878815299
// MI455X (gfx1250) — compile-verified
//
#include <hip/hip_runtime.h>
#include <cstdint>
#include <cstddef>

#define NJ   24
#define TPB  256
#define EPSF 1e-10f

// gfx1250 async global->LDS path (ASYNCcnt / CDNA5-specific), guarded so the
// host pass and older toolchains fall back to a plain coalesced float4 copy.
#if defined(__HIP_DEVICE_COMPILE__) && \
    __has_builtin(__builtin_amdgcn_global_load_async_to_lds_b128) && \
    __has_builtin(__builtin_amdgcn_s_wait_asynccnt)
#define USE_ASYNC_LDS 1
#else
#define USE_ASYNC_LDS 0
#endif

typedef int v4i __attribute__((ext_vector_type(4)));

struct Aff { float r0,r1,r2,r3,r4,r5,r6,r7,r8, px,py,pz; };

// T = T * A  (affine 3x4 compose), 27 FMA + 9 MUL
__device__ __forceinline__ void compose(Aff& T, const Aff& A) {
    float n0 = fmaf(T.r0, A.r0, fmaf(T.r1, A.r3, T.r2 * A.r6));
    float n1 = fmaf(T.r0, A.r1, fmaf(T.r1, A.r4, T.r2 * A.r7));
    float n2 = fmaf(T.r0, A.r2, fmaf(T.r1, A.r5, T.r2 * A.r8));
    float n3 = fmaf(T.r3, A.r0, fmaf(T.r4, A.r3, T.r5 * A.r6));
    float n4 = fmaf(T.r3, A.r1, fmaf(T.r4, A.r4, T.r5 * A.r7));
    float n5 = fmaf(T.r3, A.r2, fmaf(T.r4, A.r5, T.r5 * A.r8));
    float n6 = fmaf(T.r6, A.r0, fmaf(T.r7, A.r3, T.r8 * A.r6));
    float n7 = fmaf(T.r6, A.r1, fmaf(T.r7, A.r4, T.r8 * A.r7));
    float n8 = fmaf(T.r6, A.r2, fmaf(T.r7, A.r5, T.r8 * A.r8));
    float nx = fmaf(T.r0, A.px, fmaf(T.r1, A.py, fmaf(T.r2, A.pz, T.px)));
    float ny = fmaf(T.r3, A.px, fmaf(T.r4, A.py, fmaf(T.r5, A.pz, T.py)));
    float nz = fmaf(T.r6, A.px, fmaf(T.r7, A.py, fmaf(T.r8, A.pz, T.pz)));
    T.r0=n0; T.r1=n1; T.r2=n2; T.r3=n3; T.r4=n4; T.r5=n5;
    T.r6=n6; T.r7=n7; T.r8=n8; T.px=nx; T.py=ny; T.pz=nz;
}

// Build A_j(q) from precomputed per-joint constants in LDS.
// R = c*I + s*W + (1-c)*w w^T ;  p = qt*a0 + (1-c)*a1 + (qt-s)*a2
__device__ __forceinline__ Aff joint_xform(const float4* scp, int j, float qj) {
    float4 c0 = scp[j*5+0], c1 = scp[j*5+1], c2 = scp[j*5+2];
    float4 c3 = scp[j*5+3], c4 = scp[j*5+4];
    float th=c0.x, wx=c0.y, wy=c0.z, wz=c0.w;
    float oxx=c1.x, oxy=c1.y, oxz=c1.z, oyy=c1.w;
    float oyz=c2.x, ozz=c2.y, a0x=c2.z, a0y=c2.w;
    float a0z=c3.x, a1x=c3.y, a1y=c3.z, a1z=c3.w;
    float a2x=c4.x, a2y=c4.y, a2z=c4.z;

    float qt = qj * th;
    float s, c;
    __sincosf(qt, &s, &c);
    float omc = 1.0f - c;
    float qts = qt - s;

    Aff A;
    A.r0 = fmaf(omc, oxx, c);
    A.r1 = fmaf(omc, oxy, -s*wz);
    A.r2 = fmaf(omc, oxz,  s*wy);
    A.r3 = fmaf(omc, oxy,  s*wz);
    A.r4 = fmaf(omc, oyy, c);
    A.r5 = fmaf(omc, oyz, -s*wx);
    A.r6 = fmaf(omc, oxz, -s*wy);
    A.r7 = fmaf(omc, oyz,  s*wx);
    A.r8 = fmaf(omc, ozz, c);
    A.px = fmaf(qt, a0x, fmaf(omc, a1x, qts*a2x));
    A.py = fmaf(qt, a0y, fmaf(omc, a1y, qts*a2y));
    A.pz = fmaf(qt, a0z, fmaf(omc, a1z, qts*a2z));
    return A;
}

__global__ __launch_bounds__(TPB) void poe_fk_kernel(
    const float* __restrict__ qv,       // [batch, 24]
    const float* __restrict__ twist,    // [24, 6]
    const float* __restrict__ initp,    // [3]
    const float* __restrict__ initrpy,  // [3]
    float* __restrict__ out,            // [batch, 4, 4]
    int batch)
{
    __shared__ float  sq[TPB * NJ];   // 24 KB staged q tile (same layout as global)
    __shared__ float4 sc[NJ * 5];     // per-joint constants, 5 x float4 each
    __shared__ float  st[12];         // T_init affine 3x4 row-major

    const int tid = threadIdx.x;
    const int b0  = blockIdx.x * TPB;
    const float* gq = qv + (size_t)b0 * NJ;
    const bool full = (b0 + TPB <= batch);

    // ---- stage q tile into LDS, coalesced; async DMA overlaps constant setup ----
    if (full) {
#if USE_ASYNC_LDS
        #pragma unroll
        for (int k = 0; k < 6; ++k) {
            int idx = (k * TPB + tid) * 4;   // float index, 16B aligned
            __builtin_amdgcn_global_load_async_to_lds_b128(
                (__attribute__((address_space(1))) v4i*)(gq + idx),
                (__attribute__((address_space(3))) v4i*)(&sq[idx]),
                0, 0);
        }
#else
        #pragma unroll
        for (int k = 0; k < 6; ++k) {
            int idx = (k * TPB + tid) * 4;
            *(float4*)&sq[idx] = *(const float4*)(gq + idx);
        }
#endif
    } else {
        int lim = (batch - b0) * NJ;
        for (int idx = tid; idx < lim; idx += TPB) sq[idx] = gq[idx];
    }

    // ---- per-joint constants (threads 0..23) ----
    if (tid < NJ) {
        const float* tw = twist + tid * 6;
        float wx = tw[0], wy = tw[1], wz = tw[2];
        float vx = tw[3], vy = tw[4], vz = tw[5];
        float th  = sqrtf(fmaf(wx,wx, fmaf(wy,wy, wz*wz))) + EPSF;
        float inv = 1.0f / th;
        wx *= inv; wy *= inv; wz *= inv;
        vx *= inv; vy *= inv; vz *= inv;
        float cx = wy*vz - wz*vy;            // a1 = w x v
        float cy = wz*vx - wx*vz;
        float cz = wx*vy - wy*vx;
        float d  = wx*vx + wy*vy + wz*vz;    // a2 = (w.v) w - v  (= W^2 v)
        float a2x = fmaf(d, wx, -vx);
        float a2y = fmaf(d, wy, -vy);
        float a2z = fmaf(d, wz, -vz);
        sc[tid*5+0] = make_float4(th, wx, wy, wz);
        sc[tid*5+1] = make_float4(wx*wx, wx*wy, wx*wz, wy*wy);
        sc[tid*5+2] = make_float4(wy*wz, wz*wz, vx, vy);
        sc[tid*5+3] = make_float4(vz, cx, cy, cz);
        sc[tid*5+4] = make_float4(a2x, a2y, a2z, 0.0f);
    } else if (tid == NJ) {
        // T_init = pr2t(init_p, init_rpy): R = Rz(y) @ Ry(p) @ Rx(r)
        float r = initrpy[0], p = initrpy[1], y = initrpy[2];
        float sr, cr, sp, cp, sy, cy;
        __sincosf(r, &sr, &cr);
        __sincosf(p, &sp, &cp);
        __sincosf(y, &sy, &cy);
        st[0] = cy*cp; st[1] = cy*sp*sr - sy*cr; st[2]  = cy*sp*cr + sy*sr; st[3]  = initp[0];
        st[4] = sy*cp; st[5] = sy*sp*sr + cy*cr; st[6]  = sy*sp*cr - cy*sr; st[7]  = initp[1];
        st[8] = -sp;   st[9] = cp*sr;            st[10] = cp*cr;            st[11] = initp[2];
    }

#if USE_ASYNC_LDS
    if (full) __builtin_amdgcn_s_wait_asynccnt(0);
#endif
    __syncthreads();

    const int b = b0 + tid;
    if (b >= batch) return;

    // pull this element's 24 q values into registers once
    float q[NJ];
    #pragma unroll
    for (int j = 0; j < NJ; ++j) q[j] = sq[tid * NJ + j];

    // fully unrolled chain: T = A0 * A1 * ... * A23   (carry starts at I -> A0)
    Aff T = joint_xform(sc, 0, q[0]);
    #pragma unroll
    for (int j = 1; j < NJ; ++j) {
        Aff A = joint_xform(sc, j, q[j]);
        compose(T, A);
    }

    // T = T * T_init
    Aff Ti;
    Ti.r0 = st[0]; Ti.r1 = st[1]; Ti.r2 = st[2];  Ti.px = st[3];
    Ti.r3 = st[4]; Ti.r4 = st[5]; Ti.r5 = st[6];  Ti.py = st[7];
    Ti.r6 = st[8]; Ti.r7 = st[9]; Ti.r8 = st[10]; Ti.pz = st[11];
    compose(T, Ti);

    // store 4x4 row-major, 4 x b128 per element
    float4* o = (float4*)(out + (size_t)b * 16);
    o[0] = make_float4(T.r0, T.r1, T.r2, T.px);
    o[1] = make_float4(T.r3, T.r4, T.r5, T.py);
    o[2] = make_float4(T.r6, T.r7, T.r8, T.pz);
    o[3] = make_float4(0.0f, 0.0f, 0.0f, 1.0f);
}

extern "C" void kernel_launch(void* const* d_in, const int* in_sizes, int n_in,
                              void* d_out, int out_size, void* d_ws, size_t ws_size,
                              hipStream_t stream) {
    const float* qv      = (const float*)d_in[0];  // (BATCH, 24)
    const float* twist   = (const float*)d_in[1];  // (24, 6, 1)
    const float* initp   = (const float*)d_in[2];  // (1, 3, 1)
    const float* initrpy = (const float*)d_in[3];  // (1, 3, 1)
    float* out = (float*)d_out;                    // (BATCH, 4, 4)

    int batch = in_sizes[0] / NJ;
    int nblocks = (batch + TPB - 1) / TPB;
    poe_fk_kernel<<<nblocks, TPB, 0, stream>>>(qv, twist, initp, initrpy, out, batch);
}